// integralDecoder_51479478009902
// MI455X (gfx1250) — compile-verified
//
#include <hip/hip_runtime.h>

typedef __attribute__((ext_vector_type(2))) float v2f;
typedef __attribute__((ext_vector_type(8))) float v8f;

#define SAFE_B  (-1.0e30f)   // exp(anything*0 + SAFE_B) == 0.0f

// ---------------------------------------------------------------------------
// Kernel 1: per-column precompute.
//   E[i,j] = i*a[j] + b[j],  a = lp - lq,  b = (N-1)*lq - lp
//   colW[j] = trapz_weight[j] * num1[j] * denom_inv   (row-independent factor)
// The q<=0 column (p==1) is replaced by the safe triple (0, -1e30, 0) so the
// main loop needs no NaN/zero guards at all.
// ---------------------------------------------------------------------------
__global__ __launch_bounds__(256)
void integral_cols_pre(const float* __restrict__ s_in,
                       float* __restrict__ colA,
                       float* __restrict__ colB,
                       float* __restrict__ colW,
                       int two_n, float Nf)
{
    int j = blockIdx.x * blockDim.x + threadIdx.x;
    if (j >= two_n) return;

    float s  = s_in[0];
    float dx = 0.999f / (float)(two_n - 1);
    float p  = 0.001f + dx * (float)j;
    float q  = 1.0f - p;

    if (q <= 0.0f) {                 // last grid point: f == 0 there (num1==0)
        colA[j] = 0.0f;
        colB[j] = SAFE_B;
        colW[j] = 0.0f;
        return;
    }

    float lp = __logf(p);
    float lq = __logf(q);

    float num1      = 1.0f - __expf(-2.0f * Nf * s * q);
    float denom_inv = 1.0f / (1.0f - __expf(-2.0f * Nf * s));

    float w = dx;
    if (j == 0) w *= 0.5f;           // uniform-grid trapezoid endpoint

    colA[j] = lp - lq;
    colB[j] = (Nf - 1.0f) * lq - lp;
    colW[j] = w * num1 * denom_inv;
}

// ---------------------------------------------------------------------------
// Kernel 2: one block = one 16-row tile (i = 16*blockIdx.x + m), 8 waves.
// Each wave reduces its (8-aligned, padded) column strip with chained
// V_WMMA_F32_16X16X4_F32 (B = ones -> D[m,n] = C + sum_k A[m,k]).
// A-layout (32-bit A 16x4, wave32): lane<16 holds (m=lane, k=0..1),
// lane>=16 holds (m=lane-16, k=2..3)  => one v2f of g-values per lane/step.
// C/D layout: VGPR r, lanes 0-15 -> M=r; lanes 16-31 -> M=r+8.
// Two accumulators break the D->C chain so WMMA pipelines behind v_exp_f32.
// ---------------------------------------------------------------------------
__global__ __launch_bounds__(256)
void integral_rows_wmma(const float* __restrict__ colA,
                        const float* __restrict__ colB,
                        const float* __restrict__ colW,
                        const float* __restrict__ binom,
                        const float* __restrict__ u_in,
                        float* __restrict__ out,
                        int two_n, float Nf)
{
    const int tid     = threadIdx.x;
    const int nWaves  = blockDim.x >> 5;
    // per-wave strip length, 8-aligned; padded total column count
    const int per     = (((two_n + nWaves - 1) / nWaves) + 7) & ~7;
    const int stride  = per * nWaves;

    extern __shared__ float smem[];
    float* sA  = smem;
    float* sB  = smem + stride;
    float* sW  = smem + 2 * stride;
    float* red = smem + 3 * stride;              // 16 row accumulators

    // Stage column tables into LDS; pad columns get the safe triple so the
    // compute loop is completely branch-free.
    for (int j = tid; j < stride; j += blockDim.x) {
        bool in = (j < two_n);
        sA[j] = in ? colA[j] : 0.0f;
        sB[j] = in ? colB[j] : SAFE_B;
        sW[j] = in ? colW[j] : 0.0f;
    }
    if (tid < 16) red[tid] = 0.0f;
    __syncthreads();

    const int lane    = tid & 31;
    const int wave    = tid >> 5;
    const int rowBase = blockIdx.x * 16;

    const float fi   = (float)(rowBase + (lane & 15));  // row index i as float
    const int   koff = (lane >> 4) << 1;                // 0 or 2

    const int jStart = wave * per;
    const int jEnd   = jStart + per;                    // uniform per wave

    v8f acc0 = {0.f, 0.f, 0.f, 0.f, 0.f, 0.f, 0.f, 0.f};
    v8f acc1 = {0.f, 0.f, 0.f, 0.f, 0.f, 0.f, 0.f, 0.f};
    const v2f ones = {1.0f, 1.0f};

    for (int j0 = jStart; j0 < jEnd; j0 += 8) {
        const int jj = j0 + koff;                       // even -> 8B aligned
        v2f a0 = *(const v2f*)(sA + jj);
        v2f b0 = *(const v2f*)(sB + jj);
        v2f w0 = *(const v2f*)(sW + jj);
        v2f a1 = *(const v2f*)(sA + jj + 4);
        v2f b1 = *(const v2f*)(sB + jj + 4);
        v2f w1 = *(const v2f*)(sW + jj + 4);

        v2f g0 = { __expf(fmaf(fi, a0.x, b0.x)) * w0.x,
                   __expf(fmaf(fi, a0.y, b0.y)) * w0.y };
        v2f g1 = { __expf(fmaf(fi, a1.x, b1.x)) * w1.x,
                   __expf(fmaf(fi, a1.y, b1.y)) * w1.y };

        acc0 = __builtin_amdgcn_wmma_f32_16x16x4_f32(
                   false, g0, false, ones, (short)0, acc0, false, false);
        acc1 = __builtin_amdgcn_wmma_f32_16x16x4_f32(
                   false, g1, false, ones, (short)0, acc1, false, false);
    }

    v8f acc = acc0 + acc1;

    // Every column of D equals the row sums: lane 0 holds rows 0..7,
    // lane 16 holds rows 8..15 (VGPR r -> M=r / M=r+8). Combine the 8 waves.
    if (lane == 0) {
#pragma unroll
        for (int r = 0; r < 8; ++r) atomicAdd(&red[r], acc[r]);
    } else if (lane == 16) {
#pragma unroll
        for (int r = 0; r < 8; ++r) atomicAdd(&red[8 + r], acc[r]);
    }
    __syncthreads();

    if (tid < 16) {
        int i = rowBase + tid;
        if (i < two_n) {
            float scale = 4.0f * Nf * u_in[0];
            out[i] = (i == 0) ? 0.0f : scale * binom[i] * red[tid];
        }
    }
}

// ---------------------------------------------------------------------------
// Host launcher. Inputs (setup_inputs order): [0]=s (f32 scalar), [1]=N (int),
// [2]=u (f32 scalar), [3]=binomial_coef (f32[2N]). Output: f32[2N].
// ---------------------------------------------------------------------------
extern "C" void kernel_launch(void* const* d_in, const int* in_sizes, int n_in,
                              void* d_out, int out_size, void* d_ws, size_t ws_size,
                              hipStream_t stream)
{
    const float* s_in  = (const float*)d_in[0];
    const float* u_in  = (const float*)d_in[2];
    const float* binom = (const float*)d_in[3];
    float*       out   = (float*)d_out;

    const int   two_n = in_sizes[3];          // 2N (host-known via input size)
    const float Nf    = 0.5f * (float)two_n;  // N

    float* colA = (float*)d_ws;               // 3 * two_n floats of scratch
    float* colB = colA + two_n;
    float* colW = colB + two_n;

    const int T = 256;
    integral_cols_pre<<<(two_n + T - 1) / T, T, 0, stream>>>(
        s_in, colA, colB, colW, two_n, Nf);

    // Mirror the kernel's padded-stride formula for the dynamic LDS size.
    const int nWaves = T >> 5;
    const int per    = (((two_n + nWaves - 1) / nWaves) + 7) & ~7;
    const int stride = per * nWaves;
    const size_t shmem = (size_t)(3 * stride + 16) * sizeof(float);

    const int tiles = (two_n + 15) / 16;
    integral_rows_wmma<<<tiles, T, shmem, stream>>>(
        colA, colB, colW, binom, u_in, out, two_n, Nf);
}